// DAS_PW_15968688406964
// MI455X (gfx1250) — compile-verified
//
#include <hip/hip_runtime.h>
#include <math.h>
#include <stdint.h>

// ---------------- constants ----------------
static constexpr float kFS     = 20000000.0f;
static constexpr float kC      = 1540.0f;
static constexpr float kFdemod = 5000000.0f;
static constexpr float kScale  = kFS / kC;             // meters -> fractional sample index
static constexpr float kRevPerSamp = kFdemod / kFS;    // 0.25 revolutions per sample
static constexpr float kZRevK  = 2.0f * kFdemod / kC;  // revolutions per meter of gz
static constexpr float kDbPerLog2 = 6.0205999132796239f; // 20*log10(2)

#define MAX_E 256
#define MAX_A 16

// ---------------- CDNA5 (gfx1250) helpers ----------------
#if defined(__AMDGCN__) && __has_builtin(__builtin_amdgcn_global_load_async_to_lds_b32)
#define HAS_ASYNC_LDS 1
#else
#define HAS_ASYNC_LDS 0
#endif

#if HAS_ASYNC_LDS
typedef __attribute__((address_space(1))) int gint_t;   // prints as "__device__ int" in HIP
typedef __attribute__((address_space(3))) int lint_t;   // prints as "__shared__ int"
// Generic->AS1 is a bit-identical mapping for global memory on AMDGPU;
// generic LDS pointers carry the LDS offset in their low 32 bits.
__device__ __forceinline__ gint_t* as_global(const void* p) {
  return (gint_t*)(unsigned long long)(uintptr_t)p;
}
__device__ __forceinline__ lint_t* as_local(void* p) {
  return (lint_t*)(unsigned int)(uintptr_t)p;
}
#endif

__device__ __forceinline__ void wait_async_lds() {
#if HAS_ASYNC_LDS
#if __has_builtin(__builtin_amdgcn_s_wait_asynccnt)
  __builtin_amdgcn_s_wait_asynccnt(0);
#else
  asm volatile("s_wait_asynccnt 0" ::: "memory");
#endif
#endif
}

__device__ __forceinline__ float fast_sin_2pi(float r) {
#if defined(__AMDGCN__) && __has_builtin(__builtin_amdgcn_sinf)
  return __builtin_amdgcn_sinf(r);       // v_sin_f32: sin(2*pi*r)
#else
  return __sinf(r * 6.28318530717958647692f);
#endif
}
__device__ __forceinline__ float fast_cos_2pi(float r) {
#if defined(__AMDGCN__) && __has_builtin(__builtin_amdgcn_cosf)
  return __builtin_amdgcn_cosf(r);       // v_cos_f32: cos(2*pi*r)
#else
  return __cosf(r * 6.28318530717958647692f);
#endif
}
__device__ __forceinline__ float fast_log2f_(float x) {
#if defined(__AMDGCN__) && __has_builtin(__builtin_amdgcn_logf)
  return __builtin_amdgcn_logf(x);       // v_log_f32 (log2)
#else
  return __log2f(x);
#endif
}
__device__ __forceinline__ float fast_sqrtf_(float x) {
#if defined(__AMDGCN__) && __has_builtin(__builtin_amdgcn_sqrtf)
  return __builtin_amdgcn_sqrtf(x);      // v_sqrt_f32
#else
  return sqrtf(x);
#endif
}

// ---------------- kernels ----------------

// Interleave I and Q so each interpolation tap is one b64 load.
__global__ __launch_bounds__(256) void das_pack_iq(const float* __restrict__ idat,
                                                   const float* __restrict__ qdat,
                                                   float2* __restrict__ pack, int n) {
  int t = blockIdx.x * blockDim.x + threadIdx.x;
  if (t < n) pack[t] = make_float2(idat[t], qdat[t]);
}

template <bool PACKED>
__global__ __launch_bounds__(256) void das_main(const float* __restrict__ idat,
                                                const float* __restrict__ qdat,
                                                const float2* __restrict__ pack,
                                                const float* __restrict__ grid,
                                                const float* __restrict__ angles,
                                                const float* __restrict__ ele_pos,
                                                const float* __restrict__ tzero,
                                                float* __restrict__ out,
                                                float* __restrict__ blockmax,
                                                int A, int E, int S, int P) {
  __shared__ float ldsEle[3 * MAX_E];  // x, y^2 (after squaring pass), z
  __shared__ float ldsAng[MAX_A];
  __shared__ float ldsTz[MAX_A];
  __shared__ float red[256];

  const int tid = threadIdx.x;
  const int nEle = (E <= MAX_E) ? 3 * E : 0;
  const int nA   = (A <= MAX_A) ? A : 0;

#if HAS_ASYNC_LDS
  for (int k = tid; k < nEle; k += 256)
    __builtin_amdgcn_global_load_async_to_lds_b32(as_global(ele_pos + k), as_local(&ldsEle[k]), 0, 0);
  for (int k = tid; k < nA; k += 256) {
    __builtin_amdgcn_global_load_async_to_lds_b32(as_global(angles + k), as_local(&ldsAng[k]), 0, 0);
    __builtin_amdgcn_global_load_async_to_lds_b32(as_global(tzero + k),  as_local(&ldsTz[k]),  0, 0);
  }
  wait_async_lds();
#else
  for (int k = tid; k < nEle; k += 256) ldsEle[k] = ele_pos[k];
  for (int k = tid; k < nA; k += 256) { ldsAng[k] = angles[k]; ldsTz[k] = tzero[k]; }
#endif
  __syncthreads();
  if (E <= MAX_E && tid < E) {  // pre-square ey (grid y is always 0 -> dy = -ey)
    float ey = ldsEle[3 * tid + 1];
    ldsEle[3 * tid + 1] = ey * ey;
  }
  __syncthreads();

  const int p = blockIdx.x * 256 + tid;
  const bool act = (p < P);
  float gx = 0.f, gz = 0.f;
  if (act) { gx = grid[3 * p]; gz = grid[3 * p + 2]; }
  const float zrev = kZRevK * gz;  // F_DEMOD * zshift, in revolutions

  float idas = 0.f, qdas = 0.f;
  for (int a = 0; a < A; ++a) {
    float ang, tz;
    if (A <= MAX_A) { ang = ldsAng[a]; tz = ldsTz[a]; }
    else            { ang = angles[a]; tz = tzero[a]; }
    const float sa = __sinf(ang), ca = __cosf(ang);
    const float txdel = (fmaf(gx, sa, gz * ca) - tz * kC) * kScale;

    const size_t blk = (size_t)a * E * S;
    const float2* __restrict__ pbase = pack + blk;
    const float*  __restrict__ ibase = idat + blk;
    const float*  __restrict__ qbase = qdat + blk;
    __builtin_prefetch(PACKED ? (const void*)pbase : (const void*)ibase, 0, 0);

#pragma unroll 2
    for (int e = 0; e < E; ++e) {
      float ex, ey2, ez;
      if (E <= MAX_E) { ex = ldsEle[3 * e]; ey2 = ldsEle[3 * e + 1]; ez = ldsEle[3 * e + 2]; }
      else { ex = ele_pos[3 * e]; float ey = ele_pos[3 * e + 1]; ey2 = ey * ey; ez = ele_pos[3 * e + 2]; }

      const float dx = gx - ex, dz = gz - ez;
      const float r2 = fmaf(dx, dx, fmaf(dz, dz, ey2));
      const float delay = fmaf(fast_sqrtf_(r2), kScale, txdel);

      const float fi = floorf(delay);
      const float w  = delay - fi;
      const int i0 = (int)fi;
      const int i1 = i0 + 1;
      const int c0 = min(max(i0, 0), S - 1);
      const int c1 = min(max(i1, 0), S - 1);
      const bool in0 = ((unsigned)i0 < (unsigned)S);
      const bool in1 = ((unsigned)i1 < (unsigned)S);

      float iv0, iv1, qv0, qv1;
      if (PACKED) {
        const float2* __restrict__ row = pbase + (size_t)e * S;
        const float2 s0 = row[c0];
        const float2 s1 = row[c1];
        iv0 = s0.x; qv0 = s0.y; iv1 = s1.x; qv1 = s1.y;
      } else {
        const float* __restrict__ ri = ibase + (size_t)e * S;
        const float* __restrict__ rq = qbase + (size_t)e * S;
        iv0 = ri[c0]; iv1 = ri[c1]; qv0 = rq[c0]; qv1 = rq[c1];
      }
      iv0 = in0 ? iv0 : 0.f; qv0 = in0 ? qv0 : 0.f;
      iv1 = in1 ? iv1 : 0.f; qv1 = in1 ? qv1 : 0.f;

      const float ifoc = fmaf(w, iv1 - iv0, iv0);
      const float qfoc = fmaf(w, qv1 - qv0, qv0);

      // theta/2pi = 0.25*delay - (2*Fdemod/C)*gz ; exact range reduction
      float rev = fmaf(kRevPerSamp, delay, -zrev);
      rev = rev - floorf(rev);
      const float ct = fast_cos_2pi(rev);
      const float st = fast_sin_2pi(rev);

      idas = fmaf(ifoc, ct, fmaf(-qfoc, st, idas));
      qdas = fmaf(qfoc, ct, fmaf(ifoc, st, qdas));
    }
  }

  const float env = fast_sqrtf_(fmaf(idas, idas, qdas * qdas));
  const float lb  = kDbPerLog2 * fast_log2f_(env + 1e-25f);  // 20*log10(env+1e-25)
  if (act) {
    out[p]         = lb;    // bimg (normalized by a later pass)
    out[P + p]     = env;
    out[2 * P + p] = idas;
    out[3 * P + p] = qdas;
  }

  // block max of bimg
  red[tid] = act ? lb : -__builtin_inff();
  __syncthreads();
  for (int s = 128; s > 0; s >>= 1) {
    if (tid < s) red[tid] = fmaxf(red[tid], red[tid + s]);
    __syncthreads();
  }
  if (tid == 0) blockmax[blockIdx.x] = red[0];
}

__global__ __launch_bounds__(256) void das_reduce_max(const float* __restrict__ bm, int n,
                                                      float* __restrict__ gmax) {
  __shared__ float red[256];
  float m = -__builtin_inff();
  for (int k = threadIdx.x; k < n; k += 256) m = fmaxf(m, bm[k]);
  red[threadIdx.x] = m;
  __syncthreads();
  for (int s = 128; s > 0; s >>= 1) {
    if (threadIdx.x < s) red[threadIdx.x] = fmaxf(red[threadIdx.x], red[threadIdx.x + s]);
    __syncthreads();
  }
  if (threadIdx.x == 0) *gmax = red[0];
}

__global__ __launch_bounds__(256) void das_sub_max(float* __restrict__ bimg,
                                                   const float* __restrict__ gmax, int P) {
  int p = blockIdx.x * 256 + threadIdx.x;
  if (p < P) bimg[p] -= *gmax;
}

// ---------------- launcher ----------------
extern "C" void kernel_launch(void* const* d_in, const int* in_sizes, int n_in,
                              void* d_out, int out_size, void* d_ws, size_t ws_size,
                              hipStream_t stream) {
  const float* idata   = (const float*)d_in[0];
  const float* qdata   = (const float*)d_in[1];
  const float* grid    = (const float*)d_in[2];
  const float* angles  = (const float*)d_in[3];
  const float* ele_pos = (const float*)d_in[4];
  const float* tzero   = (const float*)d_in[5];

  const int A = in_sizes[3];
  const int E = in_sizes[4] / 3;
  const int S = (A * E) ? in_sizes[0] / (A * E) : 0;
  const int P = in_sizes[2] / 3;

  float* out = (float*)d_out;
  const int nblk = (P + 255) / 256;

  char* ws = (char*)d_ws;
  float* gmax     = (float*)ws;          // 1 float
  float* blockmax = (float*)(ws + 64);   // nblk floats
  size_t packOff   = ((size_t)64 + (size_t)nblk * sizeof(float) + 255) & ~(size_t)255;
  size_t packBytes = (size_t)A * E * S * sizeof(float2);
  const bool packed = (ws_size >= packOff + packBytes);
  float2* pack = (float2*)(ws + packOff);

  if (packed) {
    long n = (long)A * E * S;
    das_pack_iq<<<(int)((n + 255) / 256), 256, 0, stream>>>(idata, qdata, pack, (int)n);
    das_main<true><<<nblk, 256, 0, stream>>>(idata, qdata, pack, grid, angles, ele_pos,
                                             tzero, out, blockmax, A, E, S, P);
  } else {
    das_main<false><<<nblk, 256, 0, stream>>>(idata, qdata, pack, grid, angles, ele_pos,
                                              tzero, out, blockmax, A, E, S, P);
  }
  das_reduce_max<<<1, 256, 0, stream>>>(blockmax, nblk, gmax);
  das_sub_max<<<nblk, 256, 0, stream>>>(out, gmax, P);
}